// SpectralNetLoss_21105469292703
// MI455X (gfx1250) — compile-verified
//
#include <hip/hip_runtime.h>

typedef __attribute__((ext_vector_type(2))) float v2f;
typedef __attribute__((ext_vector_type(8))) float v8f;

#define NN 8192
#define KK 64
#define NT (NN / 16)     // 512 16x16 tiles per dimension
#define WAVES 8
#define BLK (WAVES * 32)

// ---------------------------------------------------------------------------
// Kernel 1: sq[i] = ||Y[i,:]||^2
// ---------------------------------------------------------------------------
__global__ void sqnorm_kernel(const float* __restrict__ Y, float* __restrict__ sq) {
  int i = blockIdx.x * blockDim.x + threadIdx.x;
  if (i < NN) {
    const float4* y4 = (const float4*)(Y + (size_t)i * KK);
    float s = 0.f;
#pragma unroll
    for (int k = 0; k < KK / 4; ++k) {
      float4 v = y4[k];
      s = fmaf(v.x, v.x, s);
      s = fmaf(v.y, v.y, s);
      s = fmaf(v.z, v.z, s);
      s = fmaf(v.w, v.w, s);
    }
    sq[i] = s;
  }
}

// ---------------------------------------------------------------------------
// Kernel 2: per-tile  sum( W .* max(sq_i + sq_j - 2*Y Y^T, 0) )
// One block owns one 16-row tile strip (its 8 waves split the column tiles); A-fragments (Y rows) live in VGPRs,
// B-fragments re-read per column tile (Y is 2 MB -> lives in L2/WGP$).
// W is streamed exactly once, loaded non-temporally.
// ---------------------------------------------------------------------------
__global__ void __launch_bounds__(BLK) spectral_main(
    const float* __restrict__ W, const float* __restrict__ Y,
    const float* __restrict__ sq, float* __restrict__ partials) {
  const int lane = threadIdx.x & 31;
  const int wave = threadIdx.x >> 5;
  const int ti = blockIdx.x;          // row tile index [0, 512)
  const int lo = lane & 15;
  const int hi = lane >> 4;

  // A-matrix fragments: 16x4 fp32 layout -> lane (lo) holds row M=lo,
  // VGPR pair {K = hi*2, hi*2+1} for each K-step kk.
  const int arow = ti * 16 + lo;
  v2f a[16];
#pragma unroll
  for (int kk = 0; kk < 16; ++kk) {
    a[kk] = *(const v2f*)(Y + (size_t)arow * KK + kk * 4 + hi * 2);
  }

  // sq for the 8 C rows this lane covers: M = v + 8*hi
  float sqi[8];
#pragma unroll
  for (int v = 0; v < 8; ++v) sqi[v] = sq[ti * 16 + v + 8 * hi];

  float acc = 0.f;

  for (int tj = wave; tj < NT; tj += WAVES) {
    const int brow = tj * 16 + lo;    // B holds Y^T: lane lo supplies column n=lo
    v8f c = {0.f, 0.f, 0.f, 0.f, 0.f, 0.f, 0.f, 0.f};
#pragma unroll
    for (int kk = 0; kk < 16; ++kk) {
      v2f b = *(const v2f*)(Y + (size_t)brow * KK + kk * 4 + hi * 2);
      c = __builtin_amdgcn_wmma_f32_16x16x4_f32(
          /*neg_a=*/false, a[kk], /*neg_b=*/false, b,
          /*c_mod=*/(short)0, c, /*reuse_a=*/false, /*reuse_b=*/false);
    }
    const float sqj = sq[brow];       // sq[tj*16 + N] for N = lo
    float p = 0.f;
#pragma unroll
    for (int v = 0; v < 8; ++v) {
      // C/D layout: VGPR v, lane -> element (M = v + 8*hi, N = lo)
      const int row = ti * 16 + v + 8 * hi;
      float d = sqi[v] + sqj - 2.0f * c[v];
      d = fmaxf(d, 0.f);
      const float w = __builtin_nontemporal_load(
          W + (size_t)row * NN + tj * 16 + lo);
      p = fmaf(w, d, p);
    }
    acc += p;
  }

  // wave32 reduction
#pragma unroll
  for (int off = 16; off > 0; off >>= 1) acc += __shfl_xor(acc, off, 32);

  __shared__ float red[WAVES];
  if (lane == 0) red[wave] = acc;
  __syncthreads();
  if (threadIdx.x == 0) {
    float s = 0.f;
#pragma unroll
    for (int w = 0; w < WAVES; ++w) s += red[w];
    partials[blockIdx.x] = s;
  }
}

// ---------------------------------------------------------------------------
// Kernel 3: deterministic tree reduce of 512 block partials, scale by 1/(2n)
// ---------------------------------------------------------------------------
__global__ void final_reduce(const float* __restrict__ partials,
                             float* __restrict__ out) {
  __shared__ float s[NT];
  const int t = threadIdx.x;
  s[t] = partials[t];
  __syncthreads();
  for (int off = NT / 2; off > 0; off >>= 1) {
    if (t < off) s[t] += s[t + off];
    __syncthreads();
  }
  if (t == 0) out[0] = s[0] * (1.0f / (2.0f * (float)NN));
}

// ---------------------------------------------------------------------------
extern "C" void kernel_launch(void* const* d_in, const int* in_sizes, int n_in,
                              void* d_out, int out_size, void* d_ws,
                              size_t ws_size, hipStream_t stream) {
  const float* W = (const float*)d_in[0];   // (N, N) fp32
  const float* Y = (const float*)d_in[1];   // (N, K) fp32
  float* sq = (float*)d_ws;                 // N floats
  float* partials = sq + NN;                // NT floats
  float* out = (float*)d_out;               // scalar fp32

  hipLaunchKernelGGL(sqnorm_kernel, dim3(NN / 256), dim3(256), 0, stream, Y, sq);
  hipLaunchKernelGGL(spectral_main, dim3(NT), dim3(BLK), 0, stream, W, Y, sq,
                     partials);
  hipLaunchKernelGGL(final_reduce, dim3(1), dim3(NT), 0, stream, partials, out);
}